// SimpleSAGE_9887014716200
// MI455X (gfx1250) — compile-verified
//
#include <hip/hip_runtime.h>

typedef __attribute__((ext_vector_type(2))) float v2f;
typedef __attribute__((ext_vector_type(8))) float v8f;

// ---------------------------------------------------------------------------
// Zero int buffer (for per-row edge counts; ws is poisoned 0xAA)
// ---------------------------------------------------------------------------
__global__ void zero_i_kernel(int* __restrict__ p, int n) {
    int i = blockIdx.x * blockDim.x + threadIdx.x;
    int stride = gridDim.x * blockDim.x;
    for (; i < n; i += stride) p[i] = 0;
}

// ---------------------------------------------------------------------------
// CSR build pass 1: per-row edge histogram (int atomics, cheap: E = 3.2M)
// ---------------------------------------------------------------------------
__global__ void hist_kernel(const int* __restrict__ rows, int* __restrict__ counts,
                            long long E) {
    long long e = (long long)blockIdx.x * blockDim.x + threadIdx.x;
    if (e < E) atomicAdd(&counts[rows[e]], 1);
}

// ---------------------------------------------------------------------------
// CSR build pass 2: exclusive scan of counts -> offsets[N+1], cursor copy.
// Single block of 1024 threads; chunk-per-thread + Hillis-Steele scan in LDS.
// ---------------------------------------------------------------------------
__global__ void scan_kernel(const int* __restrict__ counts, int* __restrict__ offsets,
                            int* __restrict__ cursor, int N, long long E) {
    __shared__ int sm[1024];
    const int t = threadIdx.x;
    const int chunk = (N + 1023) >> 10;
    const int base = t * chunk;

    int sum = 0;
    for (int i = 0; i < chunk; ++i) {
        int idx = base + i;
        if (idx < N) sum += counts[idx];
    }
    sm[t] = sum;
    __syncthreads();
    for (int off = 1; off < 1024; off <<= 1) {
        int v = (t >= off) ? sm[t - off] : 0;
        __syncthreads();
        sm[t] += v;
        __syncthreads();
    }
    int run = sm[t] - sum;  // exclusive prefix of this thread's chunk
    for (int i = 0; i < chunk; ++i) {
        int idx = base + i;
        if (idx < N) {
            offsets[idx] = run;
            cursor[idx]  = run;
            run += counts[idx];
        }
    }
    if (t == 0) offsets[N] = (int)E;
}

// ---------------------------------------------------------------------------
// CSR build pass 3: scatter edges into row-sorted order
// ---------------------------------------------------------------------------
__global__ void scatter_kernel(const int* __restrict__ rows, const int* __restrict__ cols,
                               const float* __restrict__ vals, int* __restrict__ cursor,
                               int* __restrict__ colsS, float* __restrict__ valsS,
                               long long E) {
    long long e = (long long)blockIdx.x * blockDim.x + threadIdx.x;
    if (e >= E) return;
    int r = rows[e];
    int p = atomicAdd(&cursor[r], 1);
    colsS[p] = cols[e];
    valsS[p] = vals[e];
}

// ---------------------------------------------------------------------------
// Row-parallel CSR SPMM: Y[r,:] = sum_e vals[e] * X[cols[e],:]
// One wave per row (avg degree 32). (col,val) pairs are loaded coalesced
// (32 edges per b32 load across lanes) and broadcast via shuffles; the source
// row is gathered with lane-parallel float4 (b128) loads from the L2-resident
// feature matrix; accumulation stays in VGPRs -> zero float atomics.
// ---------------------------------------------------------------------------
template <int D>
__global__ void spmm_csr_kernel(const int* __restrict__ offsets,
                                const int* __restrict__ colsS,
                                const float* __restrict__ valsS,
                                const float* __restrict__ X,
                                float* __restrict__ Y, int N) {
    constexpr int PER4 = D / 128;  // float4s per lane (2 for D=256, 1 for D=128)
    const int lane = threadIdx.x & 31;
    const int r = blockIdx.x * 8 + (threadIdx.x >> 5);
    if (r >= N) return;

    const int s = offsets[r];
    const int e = offsets[r + 1];

    float4 acc[PER4];
    #pragma unroll
    for (int q = 0; q < PER4; ++q) { acc[q].x = acc[q].y = acc[q].z = acc[q].w = 0.f; }

    for (int i = s; i < e; i += 32) {
        const int nb = min(32, e - i);
        int myc = 0; float myv = 0.f;
        if (lane < nb) { myc = colsS[i + lane]; myv = valsS[i + lane]; }
        for (int j = 0; j < nb; ++j) {
            const int   c = __shfl(myc, j, 32);
            const float v = __shfl(myv, j, 32);
            const float4* __restrict__ xr =
                reinterpret_cast<const float4*>(X + (size_t)c * D);
            #pragma unroll
            for (int q = 0; q < PER4; ++q) {
                const float4 xv = xr[lane + 32 * q];
                acc[q].x += v * xv.x; acc[q].y += v * xv.y;
                acc[q].z += v * xv.z; acc[q].w += v * xv.w;
            }
        }
    }

    float4* __restrict__ yr = reinterpret_cast<float4*>(Y + (size_t)r * D);
    #pragma unroll
    for (int q = 0; q < PER4; ++q) yr[lane + 32 * q] = acc[q];
}

// ---------------------------------------------------------------------------
// Dense layer with fp32 WMMA: Y = relu(A[N,K] @ W[K,128] + b)
// One WAVE owns one 16-row tile and computes ALL 8 column tiles (16x128) with
// 8 independent 16x16 f32 accumulators (64 VGPRs). Per K-step a single b64
// A-pair load feeds 8 chained V_WMMA_F32_16X16X4_F32 -> A-operand register
// reuse, A global traffic cut 8x vs wave-per-column-tile.
// Block = 256 threads = 8 waves = 8 consecutive row tiles.
// Operand layouts per CDNA5 ISA 7.12.2 (wave32):
//   A 16x4 f32 : lane m = lane&15 ; VGPR j holds K = 2*(lane>>4) + j
//   B 4x16 f32 : lane n = lane&15 ; VGPR j holds K-row = 2*(lane>>4) + j
//   C/D 16x16  : VGPR v, lanes 0-15 -> M=v, lanes 16-31 -> M=v+8 ; N = lane&15
// ---------------------------------------------------------------------------
template <int K>
__global__ void dense_relu_wmma_kernel(const float* __restrict__ A,  // [N, K]
                                       const float* __restrict__ W,  // [K, 128]
                                       const float* __restrict__ b,  // [128]
                                       float* __restrict__ Y,        // [N, 128]
                                       int N) {
    const int lane = threadIdx.x & 31;
    const int l16  = lane & 15;
    const int hi   = lane >> 4;  // 0 or 1
    const int nTiles = (N + 15) >> 4;
    const int tile = blockIdx.x * 8 + (threadIdx.x >> 5);
    if (tile >= nTiles) return;
    const int row0 = tile * 16;

    int arow = row0 + l16;
    if (arow >= N) arow = N - 1;             // clamp loads on the edge tile
    const float* __restrict__ ap = A + (size_t)arow * K;

    v8f c[8] = {};                            // 8 column tiles of 16x16 f32
    for (int kk = 0; kk < K; kk += 4) {
        const int ka = kk + 2 * hi;          // this lane's K pair
        v2f a;
        a.x = ap[ka];
        a.y = ap[ka + 1];
        const float* __restrict__ wp = W + (size_t)ka * 128 + l16;
        #pragma unroll
        for (int t = 0; t < 8; ++t) {
            v2f bm;
            bm.x = wp[t * 16];
            bm.y = wp[t * 16 + 128];
            c[t] = __builtin_amdgcn_wmma_f32_16x16x4_f32(
                       /*neg_a=*/false, a, /*neg_b=*/false, bm,
                       /*c_mod=*/(short)0, c[t], /*reuse_a=*/false, /*reuse_b=*/false);
        }
    }

    const bool full = (row0 + 16 <= N);
    #pragma unroll
    for (int t = 0; t < 8; ++t) {
        const float bias = b[t * 16 + l16];
        if (full) {                          // full tile: unguarded strided stores
            float* __restrict__ yp = Y + (size_t)(row0 + 8 * hi) * 128 + t * 16 + l16;
            #pragma unroll
            for (int v = 0; v < 8; ++v) {
                float val = c[t][v] + bias;
                yp[(size_t)v * 128] = val > 0.f ? val : 0.f;
            }
        } else {                             // ragged edge tile
            #pragma unroll
            for (int v = 0; v < 8; ++v) {
                const int m = row0 + v + 8 * hi;
                if (m < N) {
                    float val = c[t][v] + bias;
                    Y[(size_t)m * 128 + t * 16 + l16] = val > 0.f ? val : 0.f;
                }
            }
        }
    }
}

// ---------------------------------------------------------------------------
// Output projection: z[n] = dot(H[n,:], wo) + bo  (one wave per node)
// ---------------------------------------------------------------------------
__global__ void out_proj_kernel(const float* __restrict__ H,   // [N, 128]
                                const float* __restrict__ wo,  // [128]
                                const float* __restrict__ bo,  // [1]
                                float* __restrict__ z, int N) {
    const int wave = threadIdx.x >> 5;
    const int lane = threadIdx.x & 31;
    const int node = blockIdx.x * 8 + wave;
    if (node >= N) return;

    const float* __restrict__ h = H + (size_t)node * 128;
    float s = 0.f;
    #pragma unroll
    for (int i = 0; i < 4; ++i) s += h[lane + 32 * i] * wo[lane + 32 * i];
    #pragma unroll
    for (int off = 16; off > 0; off >>= 1) s += __shfl_down(s, off, 32);
    if (lane == 0) z[node] = s + bo[0];
}

// ---------------------------------------------------------------------------
// Driver
// ---------------------------------------------------------------------------
extern "C" void kernel_launch(void* const* d_in, const int* in_sizes, int n_in,
                              void* d_out, int out_size, void* d_ws, size_t ws_size,
                              hipStream_t stream) {
    const float* x     = (const float*)d_in[0];
    const int*   erow  = (const int*)  d_in[1];
    const int*   ecol  = (const int*)  d_in[2];
    const float* evals = (const float*)d_in[3];
    const float* w1    = (const float*)d_in[4];
    const float* b1    = (const float*)d_in[5];
    const float* w2    = (const float*)d_in[6];
    const float* b2    = (const float*)d_in[7];
    const float* wo    = (const float*)d_in[8];
    const float* bo    = (const float*)d_in[9];

    const int IN_DIM = 256, HID = 128;
    const long long N = in_sizes[0] / IN_DIM;
    const long long E = in_sizes[1];

    // Workspace layout in 4-byte units:
    //   counts  : [0, N)
    //   offsets : [N, 2N+1)
    //   cursor  : [2N+1, 3N+1)
    //   colsS   : [3N+1, 3N+1+E)
    //   valsS   : [3N+1+E, 3N+1+2E)
    //   feature buffers at off_h (16B aligned), peak N*384 floats:
    //     h1_pre [off_h, off_h+N*256)  (retired after dense1)
    //     h2_pre reuses [off_h, off_h+N*128)
    //     h2     reuses [off_h+N*128, off_h+N*256)
    //     h1     [off_h+N*256, off_h+N*384)
    int*   wsi     = (int*)d_ws;
    float* wsf     = (float*)d_ws;
    int*   counts  = wsi;
    int*   offsets = wsi + N;
    int*   cursor  = wsi + 2 * N + 1;
    int*   colsS   = wsi + 3 * N + 1;
    float* valsS   = wsf + 3 * N + 1 + E;
    size_t off_h   = ((size_t)(3 * N + 1 + 2 * E) + 3) & ~(size_t)3;  // 16B align
    float* h1_pre  = wsf + off_h;
    float* h2_pre  = wsf + off_h;
    float* h2      = wsf + off_h + (size_t)N * HID;
    float* h1      = wsf + off_h + (size_t)N * IN_DIM;
    float* z       = (float*)d_out;

    const int THREADS = 256;
    const int eBlocks = (int)((E + THREADS - 1) / THREADS);
    const int rBlocks = (int)((N + 7) / 8);
    const int nTiles  = (int)((N + 15) / 16);
    const int gBlocks = (nTiles + 7) / 8;    // 8 row tiles (waves) per block

    // --- build CSR (shared by both SPMM passes) ---
    zero_i_kernel<<<1024, THREADS, 0, stream>>>(counts, (int)N);
    hist_kernel<<<eBlocks, THREADS, 0, stream>>>(erow, counts, E);
    scan_kernel<<<1, 1024, 0, stream>>>(counts, offsets, cursor, (int)N, E);
    scatter_kernel<<<eBlocks, THREADS, 0, stream>>>(erow, ecol, evals, cursor,
                                                    colsS, valsS, E);

    // --- layer 1: h1 = relu(spmm(A, x) @ w1 + b1) ---
    spmm_csr_kernel<256><<<rBlocks, THREADS, 0, stream>>>(offsets, colsS, valsS,
                                                          x, h1_pre, (int)N);
    dense_relu_wmma_kernel<256><<<gBlocks, THREADS, 0, stream>>>(
        h1_pre, w1, b1, h1, (int)N);

    // --- layer 2: h2 = relu(spmm(A, h1) @ w2 + b2) ---
    spmm_csr_kernel<128><<<rBlocks, THREADS, 0, stream>>>(offsets, colsS, valsS,
                                                          h1, h2_pre, (int)N);
    dense_relu_wmma_kernel<128><<<gBlocks, THREADS, 0, stream>>>(
        h2_pre, w2, b2, h2, (int)N);

    // --- output: z = h2 @ wo + bo ---
    out_proj_kernel<<<rBlocks, THREADS, 0, stream>>>(h2, wo, bo, z, (int)N);
}